// GAT_1881195675937
// MI455X (gfx1250) — compile-verified
//
#include <hip/hip_runtime.h>
#include <hip/hip_bf16.h>
#include <math.h>

typedef float v2f __attribute__((ext_vector_type(2)));
typedef float v8f __attribute__((ext_vector_type(8)));

#define NEG_SLOPE 0.2f

// ---- order-preserving float<->uint mapping for atomic float max via u32 max ----
__device__ __forceinline__ unsigned f2mono(float f) {
    unsigned u = __float_as_uint(f);
    return (u & 0x80000000u) ? ~u : (u | 0x80000000u);
}
__device__ __forceinline__ float mono2f(unsigned u) {
    return (u & 0x80000000u) ? __uint_as_float(u & 0x7fffffffu) : __uint_as_float(~u);
}
__device__ __forceinline__ float leaky(float v) { return v > 0.f ? v : NEG_SLOPE * v; }

// ---------------- init: zero the workspace ----------------
__global__ void k_zero(float* __restrict__ p, size_t n) {
    size_t i = (size_t)blockIdx.x * blockDim.x + threadIdx.x;
    size_t stride = (size_t)gridDim.x * blockDim.x;
    for (; i < n; i += stride) p[i] = 0.f;
}

// ---------------- layer 1 GEMM: h = x @ W1 via V_WMMA_F32_16X16X4_F32 ----------------
// One wave computes a 16x16 output tile (16 nodes x 16 features), K=128 in 32 steps.
__global__ void k_gemm1(const float* __restrict__ x, const float* __restrict__ W,
                        float* __restrict__ h, int ntiles, int N) {
    int wave = (int)(((size_t)blockIdx.x * blockDim.x + threadIdx.x) >> 5);
    int lane = threadIdx.x & 31;
    if (wave >= ntiles) return;            // wave-uniform: EXEC stays all-ones for WMMA
    int m    = lane & 15;
    int half = lane >> 4;                  // 0: K pair {0,1}; 1: K pair {2,3}
    size_t row0 = (size_t)wave * 16;
    int r = (int)row0 + m; if (r >= N) r = N - 1;     // defensive clamp (N % 16 == 0 here)
    const float* xrow = x + (size_t)r * 128;

    v8f c = {};
    #pragma unroll
    for (int k0 = 0; k0 < 128; k0 += 4) {
        // A 16x4: lanes 0-15 hold (M=m, K=k0..k0+1); lanes 16-31 hold (M=m, K=k0+2..k0+3)
        v2f a = *(const v2f*)(xrow + k0 + half * 2);
        // B 4x16: VGPR0 = row K=k0 (lanes 0-15) / K=k0+2 (lanes 16-31); VGPR1 = K=k0+1 / k0+3
        v2f b;
        b.x = W[(k0 + half * 2) * 16 + m];
        b.y = W[(k0 + half * 2 + 1) * 16 + m];
        c = __builtin_amdgcn_wmma_f32_16x16x4_f32(false, a, false, b, (short)0, c, false, false);
    }
    // C/D layout: lane L, VGPR i -> row (i + 8*half), col m
    #pragma unroll
    for (int i = 0; i < 8; ++i) {
        int row = (int)row0 + i + 8 * half;
        if (row < N) h[(size_t)row * 16 + m] = c[i];
    }
}

// ---------------- per-node attention scalars: alpha = h . a ----------------
__global__ void k_alpha1(const float* __restrict__ h, const float* __restrict__ asrc,
                         const float* __restrict__ adst, float* __restrict__ as1,
                         float* __restrict__ ad1, int N) {
    int n = blockIdx.x * blockDim.x + threadIdx.x;
    if (n >= N) return;
    const float* hp = h + (size_t)n * 16;
    float s = 0.f, d = 0.f;
    #pragma unroll
    for (int i = 0; i < 16; ++i) { s += hp[i] * asrc[i]; d += hp[i] * adst[i]; }
    as1[n] = s; ad1[n] = d;
}

// ---------------- edge pass: segment max of leaky(as[s]+ad[d]) over dst ----------------
__global__ void k_edge_max(const long long* __restrict__ src, const long long* __restrict__ dst,
                           long long E, int N, const float* __restrict__ as,
                           const float* __restrict__ ad, unsigned* __restrict__ mU) {
    long long e = (long long)blockIdx.x * blockDim.x + threadIdx.x;
    long long Etot = E + N;
    if (e >= Etot) return;
    int s, d;
    if (e < E) { s = (int)src[e]; d = (int)dst[e]; } else { s = d = (int)(e - E); }
    float v = leaky(as[s] + ad[d]);
    atomicMax(mU + d, f2mono(v));
}

// ---------------- edge pass: segment sum of exp(e - m[d]) over dst ----------------
__global__ void k_edge_sum(const long long* __restrict__ src, const long long* __restrict__ dst,
                           long long E, int N, const float* __restrict__ as,
                           const float* __restrict__ ad, const unsigned* __restrict__ mU,
                           float* __restrict__ sum) {
    long long e = (long long)blockIdx.x * blockDim.x + threadIdx.x;
    long long Etot = E + N;
    if (e >= Etot) return;
    int s, d;
    if (e < E) { s = (int)src[e]; d = (int)dst[e]; } else { s = d = (int)(e - E); }
    float v = leaky(as[s] + ad[d]);
    atomicAdd(sum + d, expf(v - mono2f(mU[d])));
}

// ---------------- edge pass: agg[d,:] += h[s,:] * softmax-weight (16 lanes / edge) ----------------
__global__ void k_edge_agg16(const long long* __restrict__ src, const long long* __restrict__ dst,
                             long long E, int N, const float* __restrict__ as,
                             const float* __restrict__ ad, const unsigned* __restrict__ mU,
                             const float* __restrict__ sum, const float* __restrict__ h,
                             float* __restrict__ agg) {
    long long t = (long long)blockIdx.x * blockDim.x + threadIdx.x;
    long long total = (E + (long long)N) << 4;
    if (t >= total) return;
    long long e = t >> 4;
    int f = (int)(t & 15);
    int s, d;
    if (e < E) { s = (int)src[e]; d = (int)dst[e]; } else { s = d = (int)(e - E); }
    float v = leaky(as[s] + ad[d]);
    float a = expf(v - mono2f(mU[d])) / (sum[d] + 1e-16f);
    atomicAdd(agg + (size_t)d * 16 + f, h[(size_t)s * 16 + f] * a);
}

// ---------------- layer2 node prep: x1 = relu(agg+b1); h2 = x1.W2; alpha2 scalars ----------------
__global__ void k_node2(const float* __restrict__ agg, const float* __restrict__ b1,
                        const float* __restrict__ W2, const float* __restrict__ asrc2,
                        const float* __restrict__ adst2, float* __restrict__ h2,
                        float* __restrict__ as2, float* __restrict__ ad2, int N) {
    int n = blockIdx.x * blockDim.x + threadIdx.x;
    if (n >= N) return;
    const float* ap = agg + (size_t)n * 16;
    float acc = 0.f;
    #pragma unroll
    for (int i = 0; i < 16; ++i) {
        float v = ap[i] + b1[i];
        v = v > 0.f ? v : 0.f;
        acc += v * W2[i];
    }
    h2[n] = acc;
    as2[n] = acc * asrc2[0];
    ad2[n] = acc * adst2[0];
}

// ---------------- edge pass (scalar feature): out2[d] += h2[s] * weight ----------------
__global__ void k_edge_agg1(const long long* __restrict__ src, const long long* __restrict__ dst,
                            long long E, int N, const float* __restrict__ as,
                            const float* __restrict__ ad, const unsigned* __restrict__ mU,
                            const float* __restrict__ sum, const float* __restrict__ h2,
                            float* __restrict__ out2) {
    long long e = (long long)blockIdx.x * blockDim.x + threadIdx.x;
    long long Etot = E + N;
    if (e >= Etot) return;
    int s, d;
    if (e < E) { s = (int)src[e]; d = (int)dst[e]; } else { s = d = (int)(e - E); }
    float v = leaky(as[s] + ad[d]);
    float a = expf(v - mono2f(mU[d])) / (sum[d] + 1e-16f);
    atomicAdd(out2 + d, h2[s] * a);
}

// ---------------- final softmax over nodes ----------------
__global__ void k_zmax(const float* __restrict__ out2, const float* __restrict__ b2,
                       float* __restrict__ z, unsigned* __restrict__ maxU, int N) {
    __shared__ float red[256];
    int n = blockIdx.x * blockDim.x + threadIdx.x;
    float v = -INFINITY;
    if (n < N) { v = out2[n] + b2[0]; z[n] = v; }
    red[threadIdx.x] = v;
    __syncthreads();
    for (int off = 128; off > 0; off >>= 1) {
        if (threadIdx.x < off) red[threadIdx.x] = fmaxf(red[threadIdx.x], red[threadIdx.x + off]);
        __syncthreads();
    }
    if (threadIdx.x == 0) atomicMax(maxU, f2mono(red[0]));
}

__global__ void k_expsum(const float* __restrict__ z, const unsigned* __restrict__ maxU,
                         float* __restrict__ out, float* __restrict__ sumF, int N) {
    __shared__ float red[256];
    int n = blockIdx.x * blockDim.x + threadIdx.x;
    float ex = 0.f;
    if (n < N) { ex = expf(z[n] - mono2f(*maxU)); out[n] = ex; }
    red[threadIdx.x] = ex;
    __syncthreads();
    for (int off = 128; off > 0; off >>= 1) {
        if (threadIdx.x < off) red[threadIdx.x] += red[threadIdx.x + off];
        __syncthreads();
    }
    if (threadIdx.x == 0) atomicAdd(sumF, red[0]);
}

__global__ void k_div(float* __restrict__ out, const float* __restrict__ sumF, int N) {
    int n = blockIdx.x * blockDim.x + threadIdx.x;
    if (n < N) out[n] = out[n] / sumF[0];
}

extern "C" void kernel_launch(void* const* d_in, const int* in_sizes, int n_in,
                              void* d_out, int out_size, void* d_ws, size_t ws_size,
                              hipStream_t stream) {
    const float*     x      = (const float*)d_in[0];
    const long long* el     = (const long long*)d_in[1];
    const float*     W1     = (const float*)d_in[2];
    const float*     asrc1  = (const float*)d_in[3];
    const float*     adst1  = (const float*)d_in[4];
    const float*     b1     = (const float*)d_in[5];
    const float*     W2     = (const float*)d_in[6];
    const float*     asrc2  = (const float*)d_in[7];
    const float*     adst2  = (const float*)d_in[8];
    const float*     b2     = (const float*)d_in[9];
    float*           out    = (float*)d_out;

    const int       N    = in_sizes[0] / 128;     // 100000
    const long long E    = (long long)in_sizes[1] / 2;
    const long long Etot = E + N;
    const long long* src = el;
    const long long* dst = el + E;

    // ---- workspace layout (floats/u32, all 4-byte) ----
    float* ws   = (float*)d_ws;
    float* h1   = ws;                      // N*16
    float* agg1 = h1 + (size_t)N * 16;     // N*16
    float* as1  = agg1 + (size_t)N * 16;   // N
    float* ad1  = as1 + N;                 // N
    unsigned* m1 = (unsigned*)(ad1 + N);   // N
    float* s1   = (float*)m1 + N;          // N
    float* h2   = s1 + N;                  // N
    float* as2  = h2 + N;                  // N
    float* ad2  = as2 + N;                 // N
    unsigned* m2 = (unsigned*)(ad2 + N);   // N
    float* s2   = (float*)m2 + N;          // N
    float* out2 = s2 + N;                  // N
    float* z    = out2 + N;                // N
    unsigned* maxU = (unsigned*)(z + N);   // 1
    float* sumF = (float*)(maxU + 1);      // 1
    size_t wsElems = (size_t)N * 32 + (size_t)N * 9 + 2;

    const int B = 256;
    // 0) zero workspace (mono-mapped 0 is a valid -inf floor for the max arrays)
    k_zero<<<4096, B, 0, stream>>>(ws, wsElems);

    // 1) layer-1 GEMM via fp32 WMMA
    int ntiles = (N + 15) / 16;                         // 6250
    int gemmBlocks = (ntiles + 7) / 8;                  // 8 waves/block
    k_gemm1<<<gemmBlocks, B, 0, stream>>>(x, W1, h1, ntiles, N);

    // 2) per-node attention scalars
    int nb = (N + B - 1) / B;
    k_alpha1<<<nb, B, 0, stream>>>(h1, asrc1, adst1, as1, ad1, N);

    // 3-5) layer-1 segment softmax + aggregation
    int eb  = (int)((Etot + B - 1) / B);
    int eb16 = (int)((Etot * 16 + B - 1) / B);
    k_edge_max  <<<eb,   B, 0, stream>>>(src, dst, E, N, as1, ad1, m1);
    k_edge_sum  <<<eb,   B, 0, stream>>>(src, dst, E, N, as1, ad1, m1, s1);
    k_edge_agg16<<<eb16, B, 0, stream>>>(src, dst, E, N, as1, ad1, m1, s1, h1, agg1);

    // 6) layer-2 node prep: relu(agg+b1) @ W2 and attention scalars
    k_node2<<<nb, B, 0, stream>>>(agg1, b1, W2, asrc2, adst2, h2, as2, ad2, N);

    // 7-9) layer-2 segment softmax + scalar aggregation
    k_edge_max <<<eb, B, 0, stream>>>(src, dst, E, N, as2, ad2, m2);
    k_edge_sum <<<eb, B, 0, stream>>>(src, dst, E, N, as2, ad2, m2, s2);
    k_edge_agg1<<<eb, B, 0, stream>>>(src, dst, E, N, as2, ad2, m2, s2, h2, out2);

    // 10-12) softmax over nodes
    k_zmax  <<<nb, B, 0, stream>>>(out2, b2, z, maxU, N);
    k_expsum<<<nb, B, 0, stream>>>(z, maxU, out, sumF, N);
    k_div   <<<nb, B, 0, stream>>>(out, sumF, N);
}